// StructuredTransform_41360535060613
// MI455X (gfx1250) — compile-verified
//
#include <hip/hip_runtime.h>

typedef __attribute__((ext_vector_type(2))) float v2f;
typedef __attribute__((ext_vector_type(8))) float v8f;

#define N_RES   8192
#define N_INP   64
#define N_HAD   16384
#define N_LAYER 3
#define PITCH   140                    // 140 mod 64 = 12 -> conflict-free LDS patterns
#define LDS_BYTES (128 * PITCH * 4)    // 71,680 B (< 320 KB/WGP, 4 WG/WGP)

__device__ __forceinline__ float hsgn(int i, int j) {
  return (__popc(i & j) & 1) ? -1.0f : 1.0f;
}

__device__ __forceinline__ v8f v8f_zero() {
  v8f z;
#pragma unroll
  for (int i = 0; i < 8; ++i) z[i] = 0.0f;
  return z;
}

__global__ __launch_bounds__(256) void fwht_reservoir(
    const float* __restrict__ state, const float* __restrict__ inputs,
    const float* __restrict__ bias, const float* __restrict__ diag,
    float* __restrict__ out) {
  extern __shared__ float X[];                // logical 128x128, pitch 140
  const int row  = blockIdx.x;
  const int tid  = threadIdx.x;
  const int lane = tid & 31;
  const int wave = tid >> 5;                  // 8 waves
  const int m    = lane & 15;
  const int kh   = lane >> 4;                 // lane half selects K pair

  // ---- assemble X = diag[0] * [0.9*state | 0.4*inputs | 0] ----
  for (int n = tid; n < N_HAD; n += 256) {
    float v;
    if (n < N_RES)                   v = 0.9f * state[(size_t)row * N_RES + n];
    else if (n < N_RES + N_INP)      v = 0.4f * inputs[row * N_INP + (n - N_RES)];
    else                             v = 0.0f;
    X[(n >> 7) * PITCH + (n & 127)] = v * diag[n];
  }
  __syncthreads();

  // ---- H16 in WMMA operand layout; A- and B-layouts coincide (H symmetric) ----
  v2f Hp[4], Hm[4];
#pragma unroll
  for (int s = 0; s < 4; ++s) {
    const int j0 = 4 * s + 2 * kh;
    Hp[s].x = hsgn(m, j0);
    Hp[s].y = hsgn(m, j0 + 1);
    Hm[s].x = -Hp[s].x;
    Hm[s].y = -Hp[s].y;
  }

  for (int l = 0; l < N_LAYER; ++l) {
    // ===== Column stage: X <- H128 * X ; wave owns 16-column block =====
    {
      const int c = wave * 16 + m;            // this lane's column
      v8f acc[8];
#pragma unroll
      for (int t = 0; t < 8; ++t) acc[t] = v8f_zero();
#pragma unroll
      for (int k = 0; k < 8; ++k) {
#pragma unroll
        for (int s = 0; s < 4; ++s) {
          const int r0 = k * 16 + 4 * s + 2 * kh;        // B slice rows
          v2f xb;
          xb.x = X[r0 * PITCH + c];
          xb.y = X[(r0 + 1) * PITCH + c];
#pragma unroll
          for (int t = 0; t < 8; ++t) {                  // 8 independent acc chains
            const v2f h = (__popc(t & k) & 1) ? Hm[s] : Hp[s];
            acc[t] = __builtin_amdgcn_wmma_f32_16x16x4_f32(
                false, h, false, xb, (short)0, acc[t], false, false);
          }
        }
      }
#pragma unroll
      for (int t = 0; t < 8; ++t)
#pragma unroll
        for (int v = 0; v < 8; ++v)
          X[(t * 16 + v + 8 * kh) * PITCH + c] = acc[t][v];
    }
    __syncthreads();

    // ===== Row stage: X <- X * H128 ; wave owns 16-row block =====
    {
      const int r = wave * 16 + m;            // this lane's row (A layout)
      v8f acc[8];
#pragma unroll
      for (int t = 0; t < 8; ++t) acc[t] = v8f_zero();
#pragma unroll
      for (int k = 0; k < 8; ++k) {
#pragma unroll
        for (int s = 0; s < 4; ++s) {
          const int c0 = k * 16 + 4 * s + 2 * kh;        // A slice cols (K)
          v2f xa;
          xa.x = X[r * PITCH + c0];
          xa.y = X[r * PITCH + c0 + 1];
#pragma unroll
          for (int t = 0; t < 8; ++t) {
            const v2f h = (__popc(t & k) & 1) ? Hm[s] : Hp[s];
            acc[t] = __builtin_amdgcn_wmma_f32_16x16x4_f32(
                false, xa, false, h, (short)0, acc[t], false, false);
          }
        }
      }
#pragma unroll
      for (int t = 0; t < 8; ++t)
#pragma unroll
        for (int v = 0; v < 8; ++v)
          X[(wave * 16 + v + 8 * kh) * PITCH + t * 16 + m] = acc[t][v];
    }
    __syncthreads();

    // ---- diag for next layer ----
    if (l + 1 < N_LAYER) {
      const float* dg = diag + (size_t)(l + 1) * N_HAD;
      for (int n = tid; n < N_HAD; n += 256)
        X[(n >> 7) * PITCH + (n & 127)] *= dg[n];
      __syncthreads();
    }
  }

  // ---- epilogue: erf(X[:8192]/16384 + bias)  (ALPHA == 1) ----
  for (int n = tid; n < N_RES; n += 256) {
    const float z = X[(n >> 7) * PITCH + (n & 127)] * (1.0f / (float)N_HAD) + bias[n];
    out[(size_t)row * N_RES + n] = erff(z);
  }
}

extern "C" void kernel_launch(void* const* d_in, const int* in_sizes, int n_in,
                              void* d_out, int out_size, void* d_ws, size_t ws_size,
                              hipStream_t stream) {
  const float* state  = (const float*)d_in[0];  // [B, 8192]
  const float* inputs = (const float*)d_in[1];  // [B, 64]
  const float* bias   = (const float*)d_in[2];  // [8192]
  const float* diag   = (const float*)d_in[3];  // [3, 16384]
  float* out = (float*)d_out;                   // [B, 8192]

  const int batch = in_sizes[0] / N_RES;

  // LDS request exceeds the 64 KB default dynamic-shared cap; raise it.
  (void)hipFuncSetAttribute(reinterpret_cast<const void*>(fwht_reservoir),
                            hipFuncAttributeMaxDynamicSharedMemorySize, LDS_BYTES);

  fwht_reservoir<<<dim3(batch), dim3(256), LDS_BYTES, stream>>>(
      state, inputs, bias, diag, out);
}